// LightGCN_71717363909151
// MI455X (gfx1250) — compile-verified
//
#include <hip/hip_runtime.h>
#include <hip/hip_bf16.h>

typedef __attribute__((ext_vector_type(2))) float v2f;
typedef __attribute__((ext_vector_type(8))) float v8f;

#define DIM 64

// acc = cur = concat(emb_user, emb_item); float4-vectorized (DIM=64 keeps rows aligned).
__global__ void lgcn_init_concat(const float* __restrict__ eu, const float* __restrict__ ei,
                                 float* __restrict__ cur, float* __restrict__ acc,
                                 int nu_elems, int total_elems) {
    int i = (blockIdx.x * blockDim.x + threadIdx.x) * 4;
    if (i >= total_elems) return;
    float4 v = (i < nu_elems) ? *(const float4*)(eu + i)
                              : *(const float4*)(ei + (i - nu_elems));
    *(float4*)(cur + i) = v;
    *(float4*)(acc + i) = v;
}

__global__ void lgcn_zero(float* __restrict__ p, int total_elems) {
    int i = (blockIdx.x * blockDim.x + threadIdx.x) * 4;
    if (i >= total_elems) return;
    *(float4*)(p + i) = make_float4(0.f, 0.f, 0.f, 0.f);
}

// One wave32 per edge: lanes cover the 64-dim row as float2 each.
// Gather cur[col] (coalesced 256B), scale by val, scatter-add into next[row]
// via global_atomic_add_f32 (lands in L2 — the 38MB node buffer fits in 192MB L2).
__global__ void lgcn_spmm_edges(const float* __restrict__ vals, const int* __restrict__ row,
                                const int* __restrict__ col, const float* __restrict__ x,
                                float* __restrict__ y, int n_edges) {
    long long t = (long long)blockIdx.x * blockDim.x + threadIdx.x;
    int e = (int)(t >> 5);
    if (e >= n_edges) return;
    int lane = threadIdx.x & 31;
    int r = row[e];
    int c = col[e];
    float v = vals[e];
    int d0 = lane * 2;
    const float2 xv = *(const float2*)(x + (long long)c * DIM + d0);
    float* yp = y + (long long)r * DIM + d0;
    atomicAdd(yp,     v * xv.x);
    atomicAdd(yp + 1, v * xv.y);
}

__global__ void lgcn_acc_add(float* __restrict__ acc, const float* __restrict__ nxt,
                             int total_elems) {
    int i = (blockIdx.x * blockDim.x + threadIdx.x) * 4;
    if (i >= total_elems) return;
    float4 a = *(const float4*)(acc + i);
    float4 b = *(const float4*)(nxt + i);
    a.x += b.x; a.y += b.y; a.z += b.z; a.w += b.w;
    *(float4*)(acc + i) = a;
}

// Batched dot via V_WMMA_F32_16X16X4_F32: per 16-row tile compute G = U * V^T
// by accumulating 16 chunks of K=4, then emit the diagonal.
// A-layout (16x4 f32): lane = (M | half<<4); VGPR0/1 hold K = 2*half + {0,1}.
// B (4x16) mirrors it, so each lane loads U[row][kb..kb+1] and V[row][kb..kb+1].
// D-layout (16x16 f32): lane l, vgpr v -> M = v + 8*(l>>4), N = l&15.
// Guards are wave-granular: EXEC is all ones at every WMMA (ISA requirement).
__global__ void lgcn_gamma_wmma(const float* __restrict__ acc, const int* __restrict__ users,
                                const int* __restrict__ items, float* __restrict__ out,
                                int num_users, int batch) {
    int wave = (int)((blockIdx.x * blockDim.x + threadIdx.x) >> 5);
    if (wave * 16 >= batch) return;           // whole wave exits together
    int lane = threadIdx.x & 31;
    int rowi = lane & 15;
    int half = lane >> 4;
    int bi = wave * 16 + rowi;
    int u  = users[bi];
    int it = items[bi] + num_users;
    const float* uptr = acc + (long long)u  * DIM;
    const float* vptr = acc + (long long)it * DIM;
    v8f c = {};
#pragma unroll
    for (int j = 0; j < 16; ++j) {
        int kb = 4 * j + 2 * half;
        v2f a = *(const v2f*)(uptr + kb);
        v2f b = *(const v2f*)(vptr + kb);
        // 8 args: (neg_a, A, neg_b, B, c_mod, C, reuse_a, reuse_b)
        c = __builtin_amdgcn_wmma_f32_16x16x4_f32(false, a, false, b,
                                                  (short)0, c, false, false);
    }
    const float scale = 1.0f / 16.0f;  // light_out = acc/4 on both sides of the dot
#pragma unroll
    for (int v = 0; v < 8; ++v) {
        int m = v + (half << 3);
        if (rowi == m) out[wave * 16 + m] = c[v] * scale;
    }
}

extern "C" void kernel_launch(void* const* d_in, const int* in_sizes, int n_in,
                              void* d_out, int out_size, void* d_ws, size_t ws_size,
                              hipStream_t stream) {
    const float* emb_user = (const float*)d_in[0];
    const float* emb_item = (const float*)d_in[1];
    const float* adj_vals = (const float*)d_in[2];
    const int*   adj_row  = (const int*)d_in[3];
    const int*   adj_col  = (const int*)d_in[4];
    const int*   users    = (const int*)d_in[5];
    const int*   items    = (const int*)d_in[6];
    float* out = (float*)d_out;

    const int nu_elems  = in_sizes[0];           // NUM_USERS * 64
    const int ni_elems  = in_sizes[1];           // NUM_ITEMS * 64
    const int n_edges   = in_sizes[2];
    const int batch     = in_sizes[5];
    const int num_users = nu_elems / DIM;
    const int total     = nu_elems + ni_elems;   // N * DIM

    float* bufA = (float*)d_ws;
    float* bufB = bufA + total;
    float* acc  = bufB + total;

    const int vec_blocks = (total / 4 + 255) / 256;
    lgcn_init_concat<<<vec_blocks, 256, 0, stream>>>(emb_user, emb_item, bufA, acc,
                                                     nu_elems, total);

    float* cur = bufA;
    float* nxt = bufB;
    const long long spmm_threads = (long long)n_edges * 32;
    const int spmm_blocks = (int)((spmm_threads + 255) / 256);
    for (int layer = 0; layer < 3; ++layer) {
        lgcn_zero<<<vec_blocks, 256, 0, stream>>>(nxt, total);
        lgcn_spmm_edges<<<spmm_blocks, 256, 0, stream>>>(adj_vals, adj_row, adj_col,
                                                         cur, nxt, n_edges);
        lgcn_acc_add<<<vec_blocks, 256, 0, stream>>>(acc, nxt, total);
        float* t = cur; cur = nxt; nxt = t;
    }

    const int gamma_waves  = batch / 16;                    // 512 waves
    const int gamma_blocks = (gamma_waves * 32 + 255) / 256;
    lgcn_gamma_wmma<<<gamma_blocks, 256, 0, stream>>>(acc, users, items, out,
                                                      num_users, batch);
}